// SparseAttention_59682865545231
// MI455X (gfx1250) — compile-verified
//
#include <hip/hip_runtime.h>

typedef __attribute__((ext_vector_type(2))) float v2f;
typedef __attribute__((ext_vector_type(4))) float v4f;
typedef __attribute__((ext_vector_type(8))) float v8f;

static constexpr int Bn = 4, Mn = 4096, Dn = 64, Wn = 128;

// One block per query row m (4096 blocks), one wave per batch (4 waves = 128 threads).
// The 128 gathered column indices for row m are shared by all 4 waves via LDS.
__global__ __launch_bounds__(128) void sparse_attn_wmma_kernel(
    const float* __restrict__ q,
    const float* __restrict__ k,
    const float* __restrict__ v,
    const int*   __restrict__ cidx,
    float*       __restrict__ out)
{
    __shared__ int s_idx[Wn];
    __shared__ float __attribute__((aligned(16))) s_p[Bn][Wn];  // per-wave softmax weights

    const int m    = blockIdx.x;
    const int tid  = threadIdx.x;
    const int wv   = tid >> 5;      // wave id == batch index
    const int lane = tid & 31;
    const int half = lane >> 4;     // 0: lanes 0-15, 1: lanes 16-31
    const int col  = lane & 15;

    // Stage shared gather indices (blockDim.x == Wn == 128)
    s_idx[tid] = cidx[m * Wn + tid];
    __syncthreads();

    const int b = wv;
    const size_t rowbase = ((size_t)b * Mn + m) * Dn;
    const float* qrow = q + rowbase;
    const float* kb   = k + (size_t)b * Mn * Dn;
    const float* vb   = v + (size_t)b * Mn * Dn;

    // Preload q in WMMA B-operand layout: B[vgpr j][lane] = q[4c + j + 2*half]
    v2f qv[16];
#pragma unroll
    for (int c = 0; c < 16; ++c)
        qv[c] = *(const v2f*)(qrow + 4 * c + 2 * half);

    // ---------------- QK^T logits via V_WMMA_F32_16X16X4_F32 ----------------
    // A[vgpr j][lane] = Kg[w0 + col, 4c + j + 2*half]  (one v2f load per chunk)
    // C[vgpr r][lane] = logit[w0 + r + 8*half]         (replicated across col)
#pragma unroll
    for (int t = 0; t < 8; ++t) {
        const int kw = s_idx[t * 16 + col];
        const float* krow = kb + (size_t)kw * Dn;
        v8f acc = {};
#pragma unroll
        for (int c = 0; c < 16; ++c) {
            v2f a = *(const v2f*)(krow + 4 * c + 2 * half);
            acc = __builtin_amdgcn_wmma_f32_16x16x4_f32(
                false, a, false, qv[c], (short)0, acc, false, false);
        }
        if (col == 0) {
#pragma unroll
            for (int r = 0; r < 8; ++r)
                s_p[wv][t * 16 + 8 * half + r] = acc[r];
        }
    }
    asm volatile("s_wait_dscnt 0" ::: "memory");

    // ---------------- softmax over 128 logits (wave-private) ----------------
    v4f x = *(const v4f*)&s_p[wv][lane * 4];
    float mx = fmaxf(fmaxf(x.x, x.y), fmaxf(x.z, x.w));
#pragma unroll
    for (int off = 16; off >= 1; off >>= 1)
        mx = fmaxf(mx, __shfl_xor(mx, off, 32));
    v4f e;
    e.x = __expf(x.x - mx);
    e.y = __expf(x.y - mx);
    e.z = __expf(x.z - mx);
    e.w = __expf(x.w - mx);
    float s = (e.x + e.y) + (e.z + e.w);
#pragma unroll
    for (int off = 16; off >= 1; off >>= 1)
        s += __shfl_xor(s, off, 32);
    const float inv = 1.0f / s;
    e.x *= inv; e.y *= inv; e.z *= inv; e.w *= inv;
    *(v4f*)&s_p[wv][lane * 4] = e;
    asm volatile("s_wait_dscnt 0" ::: "memory");

    // ---------------- P·V via V_WMMA_F32_16X16X4_F32 ----------------
    // A[vgpr j][lane] = p[w0 + j + 2*half]        (ds_load_b64, bcast over M)
    // B[vgpr j][lane] = Vg[w0 + j + 2*half, d0+col] (coalesced 16-lane reads)
    // C[vgpr 0][lane] = out[d0 + col]             (replicated across M rows)
    v8f o0 = {}, o1 = {}, o2 = {}, o3 = {};
#pragma unroll 4
    for (int wc = 0; wc < 32; ++wc) {
        const int w0 = wc * 4;
        v2f pa = *(const v2f*)&s_p[wv][w0 + 2 * half];
        const int i0 = s_idx[w0 + 2 * half];
        const int i1 = s_idx[w0 + 2 * half + 1];
        const float* vr0 = vb + (size_t)i0 * Dn + col;
        const float* vr1 = vb + (size_t)i1 * Dn + col;
        v2f bv;
        bv.x = vr0[0];  bv.y = vr1[0];
        o0 = __builtin_amdgcn_wmma_f32_16x16x4_f32(false, pa, false, bv, (short)0, o0, false, false);
        bv.x = vr0[16]; bv.y = vr1[16];
        o1 = __builtin_amdgcn_wmma_f32_16x16x4_f32(false, pa, false, bv, (short)0, o1, false, false);
        bv.x = vr0[32]; bv.y = vr1[32];
        o2 = __builtin_amdgcn_wmma_f32_16x16x4_f32(false, pa, false, bv, (short)0, o2, false, false);
        bv.x = vr0[48]; bv.y = vr1[48];
        o3 = __builtin_amdgcn_wmma_f32_16x16x4_f32(false, pa, false, bv, (short)0, o3, false, false);
    }

    if (lane < 16) {
        float* orow = out + rowbase;
        orow[col]      = o0[0];
        orow[16 + col] = o1[0];
        orow[32 + col] = o2[0];
        orow[48 + col] = o3[0];
    }
}

extern "C" void kernel_launch(void* const* d_in, const int* in_sizes, int n_in,
                              void* d_out, int out_size, void* d_ws, size_t ws_size,
                              hipStream_t stream) {
    (void)in_sizes; (void)n_in; (void)out_size; (void)d_ws; (void)ws_size;
    const float* q  = (const float*)d_in[0];
    const float* k  = (const float*)d_in[1];
    const float* v  = (const float*)d_in[2];
    const int*   ci = (const int*)d_in[3];
    float* out = (float*)d_out;

    dim3 grid(Mn);
    dim3 block(Bn * 32);   // 4 waves (wave32), one per batch
    hipLaunchKernelGGL(sparse_attn_wmma_kernel, grid, block, 0, stream,
                       q, k, v, ci, out);
}